// DeformableAttention_9268539425158
// MI455X (gfx1250) — compile-verified
//
#include <hip/hip_runtime.h>

#define HH 64
#define WW 64
#define HW 4096
#define CC 256
#define NB 2
#define EPSV 1e-5f

typedef __attribute__((ext_vector_type(2))) float v2f;
typedef __attribute__((ext_vector_type(8))) float v8f;
typedef __attribute__((ext_vector_type(4))) unsigned int v4u;
typedef __attribute__((ext_vector_type(8))) int v8i;
typedef __attribute__((ext_vector_type(4))) int v4i;

// D = A(16x4) * B(4x16) + C, fp32 throughout (exact reference semantics)
__device__ __forceinline__ v8f wmma4(v2f a, v2f b, v8f c) {
  return __builtin_amdgcn_wmma_f32_16x16x4_f32(
      /*neg_a=*/false, a, /*neg_b=*/false, b,
      /*c_mod=*/(short)0, c, /*reuse_a=*/false, /*reuse_b=*/false);
}

// LDS byte offset of a __shared__ object: generic-pointer low 32 bits are the
// workgroup-relative LDS offset (ISA 10.2 aperture mapping).
__device__ __forceinline__ unsigned int lds_off(const void* p) {
  return (unsigned int)(uintptr_t)p;
}

// 2-D Tensor-Data-Mover load: tile_d0 (contiguous) x tile_d1 rows with row
// stride d0_stride (all in 4-byte elements), per D# layout in CDNA5 ISA.
// This toolchain exposes the 6-arg builtin: (g0, g1, g2, g3, g_extra, cpol).
__device__ __forceinline__ void tdm_load_2d(unsigned int ldsoff, const void* gptr,
                                            unsigned int tensor_d0, unsigned int tensor_d1,
                                            unsigned int tile_d0, unsigned int tile_d1,
                                            unsigned int d0_stride) {
  const unsigned long long ga = (unsigned long long)(uintptr_t)gptr;
  v4u g0;
  g0[0] = 1u;                                                  // count=1, user D#
  g0[1] = ldsoff;                                              // lds_addr
  g0[2] = (unsigned int)(ga & 0xffffffffull);                  // global_addr[31:0]
  g0[3] = (unsigned int)((ga >> 32) & 0x01ffffffull) | (2u << 30);  // addr[56:32], type=2
  v8i g1;
  g1[0] = (int)(2u << 16);                                     // data_size=4B, no mask
  g1[1] = (int)((tensor_d0 & 0xffffu) << 16);                  // tensor_dim0[15:0]
  g1[2] = (int)((tensor_d0 >> 16) | ((tensor_d1 & 0xffffu) << 16));
  g1[3] = (int)((tensor_d1 >> 16) | (tile_d0 << 16));          // tile_dim0
  g1[4] = (int)tile_d1;                                        // tile_dim1, tile_dim2=0
  g1[5] = (int)d0_stride;                                      // tensor_dim0_stride[31:0]
  g1[6] = 0;
  g1[7] = 0;
  const v4i z4 = {0, 0, 0, 0};                                 // groups 2/3 unused (2-D)
  const v8i z8 = {0, 0, 0, 0, 0, 0, 0, 0};
  __builtin_amdgcn_tensor_load_to_lds(g0, g1, z4, z4, z8, 0);
}

// ---------------------------------------------------------------------------
// Kernel 0: repack conv weights [O][C][3][3] -> [tap][96][256] so conv
// A-fragments are contiguous float2 in K.
// ---------------------------------------------------------------------------
__global__ __launch_bounds__(256) void k_wprep(const float* __restrict__ off_w,
                                               const float* __restrict__ at_w,
                                               float* __restrict__ wt) {
  const int i = blockIdx.x * 256 + threadIdx.x;    // < 9*96*256 = 221184
  const int tap = i / 24576;
  const int r = i % 24576;
  const int oc = r >> 8;
  const int c = r & 255;
  const float v = (oc < 64) ? off_w[((size_t)oc * 256 + c) * 9 + tap]
                            : at_w[((size_t)(oc - 64) * 256 + c) * 9 + tap];
  wt[i] = v;
}

// ---------------------------------------------------------------------------
// Kernel 1: offset conv (64ch) + attn conv (32ch) as implicit GEMM via WMMA.
// One 192-thread block (6 waves = 6 m-tiles) per 16-pixel tile; the 256x3x18
// x-halo is staged once in LDS (zero-padded borders), shared by all waves.
// Fused bias + BN; attn tiles get in-register softmax over PTS.
// ---------------------------------------------------------------------------
__global__ __launch_bounds__(192) void k_conv(
    const float* __restrict__ xin, const float* __restrict__ wt,
    const float* __restrict__ off_b, const float* __restrict__ off_g,
    const float* __restrict__ off_beta, const float* __restrict__ off_m,
    const float* __restrict__ off_v,
    const float* __restrict__ at_b, const float* __restrict__ at_g,
    const float* __restrict__ at_beta, const float* __restrict__ at_m,
    const float* __restrict__ at_v,
    float* __restrict__ off_out, float* __restrict__ attn_out) {
  __shared__ float sx[256 * 54];                 // [c][3 rows][18 cols]
  const int tid = threadIdx.x;
  const int pixflat = blockIdx.x * 16;
  const int b = pixflat >> 12;
  const int inpix = pixflat & (HW - 1);
  const int y = inpix >> 6;
  const int xb0 = inpix & 63;                    // multiple of 16

  // cooperative halo staging (exact: borders zeroed)
  for (int i = tid; i < 256 * 54; i += 192) {
    const int c = i / 54, r = i % 54;
    const int j = r / 18, k = r % 18;
    const int row = y - 1 + j;
    const int col = xb0 - 1 + k;
    const bool ok = (row >= 0) && (row < HH) && (col >= 0) && (col < WW);
    sx[i] = ok ? xin[(size_t)(b * CC + c) * HW + row * WW + col] : 0.f;
  }
  __syncthreads();

  const int lane = tid & 31;
  const int nn = lane & 15;                      // A: M row / B: N col
  const int hv = lane >> 4;                      // K half selector
  const int mtile = tid >> 5;                    // 0..5
  const int mbase = mtile * 16;

  v8f acc = {};
  for (int tap = 0; tap < 9; ++tap) {
    const int ky = tap / 3, kx = tap % 3;
    const float* wrow = wt + (size_t)tap * 24576 + (size_t)(mbase + nn) * 256;
    const float* bbase = sx + ky * 18 + kx + nn; // + c*54
    for (int kc = 0; kc < CC; kc += 4) {
      const int c0 = kc + 2 * hv;
      const v2f a = *reinterpret_cast<const v2f*>(wrow + c0);
      v2f bb;
      bb[0] = bbase[(size_t)c0 * 54];
      bb[1] = bbase[(size_t)(c0 + 1) * 54];
      acc = wmma4(a, bb, acc);
    }
  }

  // epilogue: bias + BN for channel m = mbase + r + 8*hv
  const bool isAttn = (mtile >= 4);
  float vals[8];
#pragma unroll
  for (int r = 0; r < 8; ++r) {
    const int m = mbase + r + 8 * hv;
    const int ch = isAttn ? (m - 64) : m;
    const float* cb = isAttn ? at_b : off_b;
    const float* g = isAttn ? at_g : off_g;
    const float* bt = isAttn ? at_beta : off_beta;
    const float* mn = isAttn ? at_m : off_m;
    const float* vr = isAttn ? at_v : off_v;
    const float inv = g[ch] * rsqrtf(vr[ch] + EPSV);
    vals[r] = (acc[r] + cb[ch]) * inv + (bt[ch] - mn[ch] * inv);
  }
  const int ocol = inpix + nn;
  if (!isAttn) {
#pragma unroll
    for (int r = 0; r < 8; ++r) {
      const int ch = mbase + r + 8 * hv;
      off_out[(size_t)(b * 64 + ch) * HW + ocol] = vals[r];
    }
  } else {
    // softmax over PTS: reg groups {0..3}/{4..7} each hold one head's 4 pts
#pragma unroll
    for (int g0 = 0; g0 < 8; g0 += 4) {
      float mx = vals[g0];
#pragma unroll
      for (int k = 1; k < 4; ++k) mx = fmaxf(mx, vals[g0 + k]);
      float e[4], s = 0.f;
#pragma unroll
      for (int k = 0; k < 4; ++k) { e[k] = __expf(vals[g0 + k] - mx); s += e[k]; }
      const float rs = 1.f / s;
#pragma unroll
      for (int k = 0; k < 4; ++k) {
        const int a = (mbase - 64) + g0 + k + 8 * hv;
        attn_out[(size_t)(b * 32 + a) * HW + ocol] = e[k] * rs;
      }
    }
  }
}

// ---------------------------------------------------------------------------
// Kernel 2a: per (pixel, sample) compute 4 corner indices + combined weights
// (bilinear weight * validity * softmax prob). align_corners=True semantics.
// ---------------------------------------------------------------------------
__global__ __launch_bounds__(256) void k_grid(
    const float* __restrict__ off, const float* __restrict__ attn,
    int* __restrict__ sidx, float* __restrict__ swt) {
  const int tid = blockIdx.x * 256 + threadIdx.x;   // 262144
  const int s = tid & 31;                           // h*4+p
  const int pg = tid >> 5;                          // global pixel 0..8191
  const int b = pg >> 12, inpix = pg & (HW - 1);
  const int yy = inpix >> 6, xx = inpix & 63;
  const float ox = off[(size_t)(b * 64 + 2 * s) * HW + inpix];
  const float oy = off[(size_t)(b * 64 + 2 * s + 1) * HW + inpix];
  const float aw = attn[(size_t)(b * 32 + s) * HW + inpix];
  const float scl = 63.0f / 64.0f;
  const float ix = ((float)xx + ox) * scl;
  const float iy = ((float)yy + oy) * scl;
  const float x0f = floorf(ix), y0f = floorf(iy);
  const float wx1 = ix - x0f, wx0 = 1.f - wx1;
  const float wy1 = iy - y0f, wy0 = 1.f - wy1;
  const float cx[2] = {x0f, x0f + 1.f}, cy[2] = {y0f, y0f + 1.f};
  const float wxa[2] = {wx0, wx1}, wya[2] = {wy0, wy1};
  const int base = (pg * 32 + s) * 4;
#pragma unroll
  for (int j = 0; j < 4; ++j) {
    const int kxi = j & 1, kyi = j >> 1;
    const float fx = cx[kxi], fy = cy[kyi];
    const bool val = (fx >= 0.f) && (fx <= 63.f) && (fy >= 0.f) && (fy <= 63.f);
    const int xi = (int)fminf(fmaxf(fx, 0.f), 63.f);
    const int yi = (int)fminf(fmaxf(fy, 0.f), 63.f);
    sidx[base + j] = yi * WW + xi;
    swt[base + j] = val ? (wxa[kxi] * wya[kyi] * aw) : 0.f;
  }
}

// ---------------------------------------------------------------------------
// Kernel 2b: feat[b,c,pix] = 128 weighted gathers; (idx,wt) staged in LDS,
// lanes = channels, x is L2-resident.
// ---------------------------------------------------------------------------
__global__ __launch_bounds__(256) void k_sample(
    const float* __restrict__ xin, const int* __restrict__ sidx,
    const float* __restrict__ swt, float* __restrict__ feat) {
  __shared__ int li[128];
  __shared__ float lw[128];
  const int pg = blockIdx.x;
  const int b = pg >> 12, inpix = pg & (HW - 1);
  const int t = threadIdx.x;
  if (t < 128) li[t] = sidx[(size_t)pg * 128 + t];
  else         lw[t - 128] = swt[(size_t)pg * 128 + (t - 128)];
  __syncthreads();
  const float* xb = xin + (size_t)(b * CC + t) * HW;
  float acc = 0.f;
#pragma unroll 4
  for (int j = 0; j < 128; ++j) acc += lw[j] * xb[li[j]];
  feat[(size_t)(b * CC + t) * HW + inpix] = acc;
}

// ---------------------------------------------------------------------------
// Kernel 3: fused proj GEMM + res GEMM. B tiles (feat, x: 256x16 each) are
// DMA'd once per block into LDS by the Tensor Data Mover (wave 0 issues two
// 2-D descriptors, s_wait_tensorcnt 0, barrier) -- removes the 16x redundant
// global reads across the 16 m-tile waves. Then dual f32-WMMA accumulators,
// LayerNorm over channels (shfl_xor 16 + LDS cross-wave reduce), residual+BN.
// ---------------------------------------------------------------------------
__global__ __launch_bounds__(512) void k_proj(
    const float* __restrict__ xin, const float* __restrict__ feat,
    const float* __restrict__ pw, const float* __restrict__ pb,
    const float* __restrict__ lng, const float* __restrict__ lnb,
    const float* __restrict__ rw, const float* __restrict__ rb,
    const float* __restrict__ bng, const float* __restrict__ bnb,
    const float* __restrict__ bnm, const float* __restrict__ bnv,
    float* __restrict__ out) {
  __shared__ float s_bf[CC * 16];   // feat tile [c][16 px]
  __shared__ float s_bx[CC * 16];   // x tile    [c][16 px]
  __shared__ float s_sum[16][16];
  __shared__ float s_ssq[16][16];
  const int lane = threadIdx.x & 31;
  const int wv = threadIdx.x >> 5;               // m-tile 0..15
  const int nn = lane & 15, hv = lane >> 4;
  const int pixflat = blockIdx.x * 16;
  const int b = pixflat >> 12, inpix = pixflat & (HW - 1);
  const int mbase = wv * 16;

  if (threadIdx.x < 32) {                        // wave 0 drives the TDM
    tdm_load_2d(lds_off(s_bf), feat + (size_t)(b * CC) * HW + inpix,
                /*tensor_d0=*/HW, /*tensor_d1=*/CC, /*tile_d0=*/16,
                /*tile_d1=*/CC, /*d0_stride=*/HW);
    tdm_load_2d(lds_off(s_bx), xin + (size_t)(b * CC) * HW + inpix,
                HW, CC, 16, CC, HW);
    __builtin_amdgcn_s_wait_tensorcnt(0);
  }
  __syncthreads();

  const float* pA = pw + (size_t)(mbase + nn) * CC;
  const float* rA = rw + (size_t)(mbase + nn) * CC;

  v8f aO = {}, aR = {};
  for (int k = 0; k < CC; k += 4) {
    const int c0 = k + 2 * hv;
    const v2f ap = *reinterpret_cast<const v2f*>(pA + c0);
    const v2f ar = *reinterpret_cast<const v2f*>(rA + c0);
    v2f bf, bx;
    bf[0] = s_bf[c0 * 16 + nn];
    bf[1] = s_bf[(c0 + 1) * 16 + nn];
    bx[0] = s_bx[c0 * 16 + nn];
    bx[1] = s_bx[(c0 + 1) * 16 + nn];
    aO = wmma4(ap, bf, aO);
    aR = wmma4(ar, bx, aR);
  }

  float o8[8], r8[8], ps = 0.f, pq = 0.f;
#pragma unroll
  for (int r = 0; r < 8; ++r) {
    const int o = mbase + r + 8 * hv;
    const float ov = aO[r] + pb[o];
    o8[r] = ov;
    r8[r] = aR[r] + rb[o];
    ps += ov; pq += ov * ov;
  }
  ps += __shfl_xor(ps, 16, 32);
  pq += __shfl_xor(pq, 16, 32);
  if (hv == 0) { s_sum[wv][nn] = ps; s_ssq[wv][nn] = pq; }
  __syncthreads();
  float tot = 0.f, totq = 0.f;
#pragma unroll
  for (int w = 0; w < 16; ++w) { tot += s_sum[w][nn]; totq += s_ssq[w][nn]; }
  const float mu = tot * (1.f / 256.f);
  const float var = totq * (1.f / 256.f) - mu * mu;
  const float rinv = rsqrtf(var + EPSV);
#pragma unroll
  for (int r = 0; r < 8; ++r) {
    const int o = mbase + r + 8 * hv;
    const float ln = (o8[r] - mu) * rinv * lng[o] + lnb[o];
    const float inv = bng[o] * rsqrtf(bnv[o] + EPSV);
    out[(size_t)(b * CC + o) * HW + inpix + nn] =
        (ln + r8[r]) * inv + (bnb[o] - bnm[o] * inv);
  }
}

extern "C" void kernel_launch(void* const* d_in, const int* in_sizes, int n_in,
                              void* d_out, int out_size, void* d_ws, size_t ws_size,
                              hipStream_t stream) {
  const float* x      = (const float*)d_in[0];
  const float* off_w  = (const float*)d_in[1];
  const float* off_b  = (const float*)d_in[2];
  const float* off_g  = (const float*)d_in[3];
  const float* off_bt = (const float*)d_in[4];
  const float* off_m  = (const float*)d_in[5];
  const float* off_v  = (const float*)d_in[6];
  const float* at_w   = (const float*)d_in[7];
  const float* at_b   = (const float*)d_in[8];
  const float* at_g   = (const float*)d_in[9];
  const float* at_bt  = (const float*)d_in[10];
  const float* at_m   = (const float*)d_in[11];
  const float* at_v   = (const float*)d_in[12];
  const float* pw     = (const float*)d_in[13];
  const float* pb     = (const float*)d_in[14];
  const float* lng    = (const float*)d_in[15];
  const float* lnb    = (const float*)d_in[16];
  const float* rw     = (const float*)d_in[17];
  const float* rb     = (const float*)d_in[18];
  const float* bng    = (const float*)d_in[19];
  const float* bnb    = (const float*)d_in[20];
  const float* bnm    = (const float*)d_in[21];
  const float* bnv    = (const float*)d_in[22];

  // workspace layout (~21 MB)
  float* wt       = (float*)d_ws;                              // 9*96*256
  float* off_buf  = wt + (size_t)9 * 96 * 256;                 // 2*64*4096
  float* attn_buf = off_buf + (size_t)NB * 64 * HW;            // 2*32*4096
  int*   sidx     = (int*)(attn_buf + (size_t)NB * 32 * HW);   // 8192*128
  float* swt      = (float*)(sidx + (size_t)8192 * 128);       // 8192*128
  float* feat     = swt + (size_t)8192 * 128;                  // 2*256*4096

  k_wprep<<<864, 256, 0, stream>>>(off_w, at_w, wt);
  k_conv<<<512, 192, 0, stream>>>(x, wt, off_b, off_g, off_bt, off_m, off_v,
                                  at_b, at_g, at_bt, at_m, at_v,
                                  off_buf, attn_buf);
  k_grid<<<1024, 256, 0, stream>>>(off_buf, attn_buf, sidx, swt);
  k_sample<<<8192, 256, 0, stream>>>(x, sidx, swt, feat);
  k_proj<<<512, 512, 0, stream>>>(x, feat, pw, pb, lng, lnb, rw, rb,
                                  bng, bnb, bnm, bnv, (float*)d_out);
}